// SSIMLoss_10273561772296
// MI455X (gfx1250) — compile-verified
//
#include <hip/hip_runtime.h>

typedef __attribute__((ext_vector_type(2))) float v2f;
typedef __attribute__((ext_vector_type(8))) float v8f;

#define IMG    512
#define TILE   16
#define PATCH  26      // TILE + 10 halo
#define PROWS  32      // patch rows padded so rt=1 A-reads stay in-bounds
#define PSTR   28      // patch cols padded to K=28 (7 * 4)
#define KSTEPS 7
#define NMAPS  5       // x, y, x^2, y^2, x*y

__device__ __forceinline__ v8f wmma4(v2f a, v2f b, v8f c) {
  // D(16x16,f32) = A(16x4,f32) * B(4x16,f32) + C   -> v_wmma_f32_16x16x4_f32
  return __builtin_amdgcn_wmma_f32_16x16x4_f32(
      /*neg_a=*/false, a, /*neg_b=*/false, b,
      /*c_mod=*/(short)0, c, /*reuse_a=*/false, /*reuse_b=*/false);
}

__global__ __launch_bounds__(32)
void ssim_tile_kernel(const float* __restrict__ pred,
                      const float* __restrict__ targ,
                      float* __restrict__ partial)
{
  __shared__ float sP[PROWS * PSTR];
  __shared__ float sT[PROWS * PSTR];
  __shared__ float sH[NMAPS][28 * 16];   // horizontal-pass result, K padded to 28
  __shared__ float sBand[48];            // sBand[d+15] = g[d] for d in [0,10], else 0

  const int lane = threadIdx.x;

  // Build normalized Gaussian band table (matches reference: exp(-d^2/(2*1.5^2)), normalized)
  if (lane < 43) {
    int d = lane - 15;
    float v = 0.f;
    if (d >= 0 && d <= 10) {
      float s = 0.f, gv = 0.f;
      for (int i = 0; i < 11; ++i) {
        float x = (float)(i - 5);
        float e = expf(-x * x / 4.5f);
        s += e;
        if (i == d) gv = e;
      }
      v = gv / s;
    }
    sBand[lane] = v;
  }

  // Zero patches (covers halo padding + padded rows/cols)
  for (int i = lane; i < PROWS * PSTR; i += 32) { sP[i] = 0.f; sT[i] = 0.f; }

  const int tile = blockIdx.x;
  const int nc   = tile >> 10;        // plane index (n*3 + c), 1024 tiles per plane
  const int ti   = tile & 1023;
  const int ty   = ti >> 5;
  const int tx   = ti & 31;
  const int row0 = ty * TILE - 5;
  const int col0 = tx * TILE - 5;
  const float* pB = pred + (size_t)nc * IMG * IMG;
  const float* tB = targ + (size_t)nc * IMG * IMG;
  __syncthreads();

  // Load 26x26 zero-padded patches (single global read of each input element)
  for (int i = lane; i < PATCH * PATCH; i += 32) {
    int r = i / PATCH, c = i - r * PATCH;
    int gr = row0 + r, gc = col0 + c;
    float pv = 0.f, tv = 0.f;
    if ((unsigned)gr < IMG && (unsigned)gc < IMG) {
      size_t o = (size_t)gr * IMG + gc;
      pv = pB[o];
      tv = tB[o];
    }
    sP[r * PSTR + c] = pv;
    sT[r * PSTR + c] = tv;
  }
  __syncthreads();

  const int nn = lane & 15;          // N (or M) index within fragment
  const int kh = (lane >> 4) << 1;   // K half-offset: lanes 16-31 hold K+2/K+3

  // Banded Gaussian fragments: serve as B for horizontal pass and A for vertical pass
  v2f gb[KSTEPS];
#pragma unroll
  for (int kk = 0; kk < KSTEPS; ++kk) {
    int k0 = kk * 4 + kh;
    gb[kk].x = sBand[k0     - nn + 15];
    gb[kk].y = sBand[k0 + 1 - nn + 15];
  }

  const v8f z8 = {0.f, 0.f, 0.f, 0.f, 0.f, 0.f, 0.f, 0.f};

  // ---- Horizontal pass: H(26x16) = Patch(26x28) x Gb(28x16), 5 maps ----
  v8f acc[NMAPS][2];
#pragma unroll
  for (int m = 0; m < NMAPS; ++m) { acc[m][0] = z8; acc[m][1] = z8; }

#pragma unroll
  for (int rt = 0; rt < 2; ++rt) {
#pragma unroll
    for (int kk = 0; kk < KSTEPS; ++kk) {
      int row = rt * 16 + nn;
      int k0  = kk * 4 + kh;
      float p0 = sP[row * PSTR + k0], p1 = sP[row * PSTR + k0 + 1];
      float t0 = sT[row * PSTR + k0], t1 = sT[row * PSTR + k0 + 1];
      v2f a;
      a.x = p0;      a.y = p1;      acc[0][rt] = wmma4(a, gb[kk], acc[0][rt]);
      a.x = t0;      a.y = t1;      acc[1][rt] = wmma4(a, gb[kk], acc[1][rt]);
      a.x = p0 * p0; a.y = p1 * p1; acc[2][rt] = wmma4(a, gb[kk], acc[2][rt]);
      a.x = t0 * t0; a.y = t1 * t1; acc[3][rt] = wmma4(a, gb[kk], acc[3][rt]);
      a.x = p0 * t0; a.y = p1 * t1; acc[4][rt] = wmma4(a, gb[kk], acc[4][rt]);
    }
  }

  // Scatter D fragments (row-striped) into sH; zero K-pad rows 26,27
  const int rowOffD = (lane >> 4) << 3;   // lanes 0-15 -> rows r, lanes 16-31 -> rows 8+r
#pragma unroll
  for (int m = 0; m < NMAPS; ++m) {
#pragma unroll
    for (int rt = 0; rt < 2; ++rt) {
#pragma unroll
      for (int r = 0; r < 8; ++r) {
        int row = rt * 16 + rowOffD + r;
        if (row < PATCH) sH[m][row * 16 + nn] = acc[m][rt][r];
      }
    }
    sH[m][26 * 16 + lane] = 0.f;   // 32 entries = rows 26 and 27
  }
  __syncthreads();

  // ---- Vertical pass: Out(16x16) = GbT(16x28) x H(28x16), 5 maps ----
  v8f vac[NMAPS];
#pragma unroll
  for (int m = 0; m < NMAPS; ++m) vac[m] = z8;

#pragma unroll
  for (int kk = 0; kk < KSTEPS; ++kk) {
    int k0 = kk * 4 + kh;
    v2f a = gb[kk];                 // A[m][k] = g[k-m] == band fragment
#pragma unroll
    for (int m = 0; m < NMAPS; ++m) {
      v2f b;
      b.x = sH[m][k0 * 16 + nn];
      b.y = sH[m][(k0 + 1) * 16 + nn];
      vac[m] = wmma4(a, b, vac[m]);
    }
  }

  // ---- Elementwise SSIM + wave reduction ----
  const float c1 = 0.01f * 0.01f;
  const float c2 = 0.03f * 0.03f;
  float lsum = 0.f;
#pragma unroll
  for (int r = 0; r < 8; ++r) {
    float mx  = vac[0][r], my  = vac[1][r];
    float ex2 = vac[2][r], ey2 = vac[3][r], exy = vac[4][r];
    float mx2 = mx * mx, my2 = my * my, mxy = mx * my;
    float sxx = ex2 - mx2, syy = ey2 - my2, sxy = exy - mxy;
    float num = (2.f * mxy + c1) * (2.f * sxy + c2);
    float den = (mx2 + my2 + c1) * (sxx + syy + c2);
    lsum += num / fmaxf(den, 1e-8f);
  }
#pragma unroll
  for (int off = 16; off > 0; off >>= 1)
    lsum += __shfl_xor(lsum, off, 32);
  if (lane == 0) partial[blockIdx.x] = lsum;
}

__global__ __launch_bounds__(256)
void ssim_reduce_kernel(const float* __restrict__ partial, int n,
                        float* __restrict__ out, float inv_count)
{
  __shared__ float sdata[256];
  float s = 0.f;
  for (int i = threadIdx.x; i < n; i += 256) s += partial[i];
  sdata[threadIdx.x] = s;
  __syncthreads();
  for (int stride = 128; stride > 0; stride >>= 1) {
    if (threadIdx.x < stride) sdata[threadIdx.x] += sdata[threadIdx.x + stride];
    __syncthreads();
  }
  if (threadIdx.x == 0) out[0] = 1.0f - sdata[0] * inv_count;
}

extern "C" void kernel_launch(void* const* d_in, const int* in_sizes, int n_in,
                              void* d_out, int out_size, void* d_ws, size_t ws_size,
                              hipStream_t stream) {
  const float* pred = (const float*)d_in[0];
  const float* targ = (const float*)d_in[1];
  float* out = (float*)d_out;
  float* ws  = (float*)d_ws;

  const int nTiles = 48 * 32 * 32;                // (N*C) * (512/16)^2 = 49152
  ssim_tile_kernel<<<nTiles, 32, 0, stream>>>(pred, targ, ws);
  ssim_reduce_kernel<<<1, 256, 0, stream>>>(ws, nTiles, out,
                                            1.0f / (16.f * 3.f * 512.f * 512.f));
}